// Merge_xs_61083024884172
// MI455X (gfx1250) — compile-verified
//
#include <hip/hip_runtime.h>
#include <math.h>

// Merge_xs: bandwidth-bound fused kernel for MI455X (gfx1250, wave32).
// One row per wave32; lane L owns cols [4L,4L+4) and [128+4L,128+4L+4).
// Cross-lane reductions: V_WMMA_F32_16X16X4_F32, two reductions per op via
// a column-selector B matrix.

typedef __attribute__((ext_vector_type(4))) float v4f;
typedef __attribute__((ext_vector_type(2))) float v2f;
typedef __attribute__((ext_vector_type(8))) float v8f;

#define DIM 256
#define NV4 (DIM / 4)  // 64 float4 per row

__device__ __forceinline__ float dot4(v4f a, v4f b) {
  return fmaf(a.x, b.x, fmaf(a.y, b.y, fmaf(a.z, b.z, a.w * b.w)));
}

// Dual 32-lane f32 sum with ONE V_WMMA_F32_16X16X4_F32.
//   A = {pa, pb}: pa fills the even-K slots (VGPR0), pb the odd-K slots.
//   B: VGPR0 (even-K rows) = [n<8], VGPR1 (odd-K rows) = [n>=8], n = lane%16.
//   => D[m,n] = (n<8) ? pa[m]+pa[m+16] : pb[m]+pb[m+16].
// Per-lane sum of the 8 D VGPRs + shfl_xor(16) completes the 16-way m-sum:
// lanes with n<8 hold sum(pa), lanes with n>=8 hold sum(pb); shfl_xor(8)
// exchanges the two so every lane ends with both totals.
struct SumPair { float a, b; };

__device__ __forceinline__ SumPair wave_sum_pair(float pa, float pb,
                                                 v2f bsel, bool nlow) {
  v2f av; av.x = pa; av.y = pb;
  v8f c = {0.f, 0.f, 0.f, 0.f, 0.f, 0.f, 0.f, 0.f};
  v8f d = __builtin_amdgcn_wmma_f32_16x16x4_f32(
      /*neg_a=*/false, av, /*neg_b=*/false, bsel,
      /*c_mod=*/(short)0, c, /*reuse_a=*/false, /*reuse_b=*/false);
  float s = ((d[0] + d[1]) + (d[2] + d[3])) + ((d[4] + d[5]) + (d[6] + d[7]));
  s += __shfl_xor(s, 16, 32);
  float t = __shfl_xor(s, 8, 32);
  SumPair r;
  r.a = nlow ? s : t;
  r.b = nlow ? t : s;
  return r;
}

__global__ __launch_bounds__(256) void merge_xs_kernel(
    const float* __restrict__ x0, const float* __restrict__ x1,
    const float* __restrict__ x2, const float* __restrict__ x3,
    const float* __restrict__ w_att, const float* __restrict__ b_att,
    float* __restrict__ out, int N) {
  const int wave = threadIdx.x >> 5;
  const int lane = threadIdx.x & 31;
  const int row = blockIdx.x * 8 + wave;
  if (row >= N) return;  // wave-uniform: EXEC stays all-ones past here

  const size_t rb = (size_t)row * NV4;
  const v4f* q4 = (const v4f*)x0 + rb;
  const v4f* a4 = (const v4f*)x1 + rb;
  const v4f* b4 = (const v4f*)x2 + rb;
  const v4f* c4 = (const v4f*)x3 + rb;
  const v4f* w4 = (const v4f*)w_att;

  // Streaming data: non-temporal 128-bit loads, 512 B contiguous per wave.
  v4f q0 = __builtin_nontemporal_load(q4 + lane);
  v4f q1 = __builtin_nontemporal_load(q4 + 32 + lane);
  v4f a0 = __builtin_nontemporal_load(a4 + lane);
  v4f a1 = __builtin_nontemporal_load(a4 + 32 + lane);
  v4f b0 = __builtin_nontemporal_load(b4 + lane);
  v4f b1 = __builtin_nontemporal_load(b4 + 32 + lane);
  v4f c0 = __builtin_nontemporal_load(c4 + lane);
  v4f c1 = __builtin_nontemporal_load(c4 + 32 + lane);
  // Weights: hot in WGP$/L2, regular temporal loads.
  v4f wm0 = w4[lane], wm1 = w4[32 + lane];
  v4f wq0 = w4[64 + lane], wq1 = w4[96 + lane];
  const float bias = b_att[0];

  // Column-selector B for the paired reductions (n = lane%16; bit3 tests n<8).
  const bool nlow = (lane & 8) == 0;
  v2f bsel; bsel.x = nlow ? 1.0f : 0.0f; bsel.y = nlow ? 0.0f : 1.0f;

  // Per-lane partials -> four dual WMMA reductions (8 exact 32-lane sums).
  SumPair rq = wave_sum_pair(dot4(q0, q0) + dot4(q1, q1),
                             dot4(q0, wq0) + dot4(q1, wq1), bsel, nlow);
  SumPair r1 = wave_sum_pair(dot4(a0, a0) + dot4(a1, a1),
                             dot4(a0, wm0) + dot4(a1, wm1), bsel, nlow);
  SumPair r2 = wave_sum_pair(dot4(b0, b0) + dot4(b1, b1),
                             dot4(b0, wm0) + dot4(b1, wm1), bsel, nlow);
  SumPair r3 = wave_sum_pair(dot4(c0, c0) + dot4(c1, c1),
                             dot4(c0, wm0) + dot4(c1, wm1), bsel, nlow);

  // l2norm with clip(norm, 1e-12); v_rcp_f32 is <=1 ulp.
  const float invq = __builtin_amdgcn_rcpf(fmaxf(sqrtf(rq.a), 1e-12f));
  const float inv1 = __builtin_amdgcn_rcpf(fmaxf(sqrtf(r1.a), 1e-12f));
  const float inv2 = __builtin_amdgcn_rcpf(fmaxf(sqrtf(r2.a), 1e-12f));
  const float inv3 = __builtin_amdgcn_rcpf(fmaxf(sqrtf(r3.a), 1e-12f));

  const float sq = fmaf(rq.b, invq, bias);
  float s0 = fmaf(r1.b, inv1, sq);
  float s1 = fmaf(r2.b, inv2, sq);
  float s2 = fmaf(r3.b, inv3, sq);
  // leaky_relu(0.01)
  s0 = s0 > 0.0f ? s0 : 0.01f * s0;
  s1 = s1 > 0.0f ? s1 : 0.01f * s1;
  s2 = s2 > 0.0f ? s2 : 0.01f * s2;
  // softmax over the 3 messages
  const float mx = fmaxf(s0, fmaxf(s1, s2));
  const float e0 = __expf(s0 - mx);
  const float e1 = __expf(s1 - mx);
  const float e2 = __expf(s2 - mx);
  const float einv = __builtin_amdgcn_rcpf(e0 + e1 + e2);
  const float p0 = e0 * einv, p1 = e1 * einv, p2 = e2 * einv;

  // embedding = x0 + sum_l p_l * (m_l / ||m_l||)
  const float k1 = p0 * inv1, k2 = p1 * inv2, k3 = p2 * inv3;
  v4f o0 = q0 + k1 * a0 + k2 * b0 + k3 * c0;
  v4f o1 = q1 + k1 * a1 + k2 * b1 + k3 * c1;

  v4f* ov = (v4f*)out + rb;
  __builtin_nontemporal_store(o0, ov + lane);
  __builtin_nontemporal_store(o1, ov + 32 + lane);

  // score output: flat (3, N) at offset N*DIM. One store from lanes 0..2
  // (divergence is after all WMMAs, so EXEC was all-ones for them).
  if (lane < 3) {
    const float pv = lane == 0 ? p0 : (lane == 1 ? p1 : p2);
    out[(size_t)N * DIM + (size_t)lane * N + row] = pv;
  }
}

extern "C" void kernel_launch(void* const* d_in, const int* in_sizes, int n_in,
                              void* d_out, int out_size, void* d_ws,
                              size_t ws_size, hipStream_t stream) {
  (void)n_in; (void)out_size; (void)d_ws; (void)ws_size;
  const float* x0 = (const float*)d_in[0];
  const float* x1 = (const float*)d_in[1];
  const float* x2 = (const float*)d_in[2];
  const float* x3 = (const float*)d_in[3];
  const float* w_att = (const float*)d_in[4];
  const float* b_att = (const float*)d_in[5];
  float* out = (float*)d_out;

  const int N = in_sizes[0] / DIM;          // 200000
  const int blocks = (N + 7) / 8;           // 8 rows (waves) per 256-thread block
  merge_xs_kernel<<<blocks, 256, 0, stream>>>(x0, x1, x2, x3, w_att, b_att,
                                              out, N);
}